// NaturalGradientDescentVelNet_53618371724063
// MI455X (gfx1250) — compile-verified
//
#include <hip/hip_runtime.h>

#define NDIM   8
#define HID    100
#define HP     104      // K padded to multiple of 4 (and of 8 for paired loads)
#define HR     112      // out-rows padded to multiple of 16
#define NT     7        // 112/16 tiles
#define WAVES  8
#define ELEMS  16       // batch elements per wave
#define SLOPE  0.01f
#define EPSV   1e-12f

typedef __attribute__((ext_vector_type(2))) float v2f;
typedef __attribute__((ext_vector_type(8))) float v8f;

// ---------------- LDS layout (float offsets) ----------------
#define OFF_W1 0            // [112 x 8]
#define OFF_V1 896          // [112 x 8]
#define OFF_W2 1792         // [112 x 104]
#define OFF_V2 13440        // [112 x 104]
#define OFF_W3 25088        // [16 x 104]
#define OFF_V3 26752        // [16 x 104]
#define OFF_B1 28416
#define OFF_B2 28528
#define OFF_C1 28640
#define OFF_C2 28752
#define OFF_B3 28864
#define OFF_C3 28872
#define OFF_WV 28880        // per-wave region start
#define WS     6304         // per-wave stride (floats)
#define W_ACT  0            // [16 x 104] activations (a1->a2->g1->g2); T aliases this
#define W_D1   1664         // [16 x 104]
#define W_D2   3328         // [16 x 104]
#define W_J    4992         // [16 x 64]
#define W_X    6016         // [16 x 8]
#define W_YD   6144         // [16 x 8]
#define W_S0   6272         // [16]
#define LDS_FLOATS (OFF_WV + WAVES*WS)   // 79312 floats = 317,248 B < 320 KB

__device__ __forceinline__ v8f v8zero() {
  v8f z;
  #pragma unroll
  for (int i = 0; i < 8; ++i) z[i] = 0.0f;
  return z;
}

// D(16x16,f32) = A(16x4,f32) * B(4x16,f32) + C  — native fp32 WMMA
__device__ __forceinline__ v8f wmma4(v2f a, v2f b, v8f c) {
  return __builtin_amdgcn_wmma_f32_16x16x4_f32(false, a, false, b, (short)0, c,
                                               false, false);
}

// ---- layer: [16 x 8] @ [8 x 100]^T  (K = 8, two WMMA steps per N tile) ----
__device__ __forceinline__ void layer_in(const float* Xb, const float* Wp,
                                         const float* bias, float* Act, float* D,
                                         int l16, int kh, int mb)
{
  const v2f* Ar = (const v2f*)(Xb + l16*NDIM + kh);
  const v2f a0 = Ar[0];          // k = kh+0..1
  const v2f a1 = Ar[2];          // k = kh+4..5
  for (int nt = 0; nt < NT; ++nt) {
    const v2f* Br = (const v2f*)(Wp + (nt*16 + l16)*NDIM + kh);
    v8f acc = v8zero();
    acc = wmma4(a0, Br[0], acc);
    acc = wmma4(a1, Br[2], acc);
    int col = nt*16 + l16;
    if (col < HP) {
      float bv  = bias[col];
      bool  pad = (col >= HID);
      #pragma unroll
      for (int r = 0; r < 8; ++r) {
        float h  = acc[r] + bv;
        float av = (h > 0.f) ? h : SLOPE * h;
        Act[(mb + r)*HP + col] = pad ? 0.f : av;
        if (D) D[(mb + r)*HP + col] = pad ? 0.f : ((h > 0.f) ? 1.f : SLOPE);
      }
    }
  }
}

// ---- layer: [16 x 100] @ [100 x 100]^T  (K padded to 104, unrolled x2) ----
__device__ __forceinline__ void layer_hid(const float* Ain, const float* Wp,
                                          const float* bias, float* Aout, float* D,
                                          int l16, int kh, int mb)
{
  v8f acc[NT];
  #pragma unroll
  for (int t = 0; t < NT; ++t) acc[t] = v8zero();
  const v2f* Ar = (const v2f*)(Ain + l16*HP + kh);
  #pragma unroll 1
  for (int k0 = 0; k0 < HP; k0 += 8) {
    const int ki = k0 >> 1;
    v2f a0 = Ar[ki];
    v2f a1 = Ar[ki + 2];
    #pragma unroll
    for (int t = 0; t < NT; ++t) {
      const v2f* Br = (const v2f*)(Wp + (t*16 + l16)*HP + kh);
      acc[t] = wmma4(a0, Br[ki],     acc[t]);
      acc[t] = wmma4(a1, Br[ki + 2], acc[t]);
    }
  }
  #pragma unroll
  for (int t = 0; t < NT; ++t) {
    int col = t*16 + l16;
    if (col < HP) {
      float bv  = bias[col];
      bool  pad = (col >= HID);
      #pragma unroll
      for (int r = 0; r < 8; ++r) {
        float h  = acc[t][r] + bv;
        float av = (h > 0.f) ? h : SLOPE * h;
        Aout[(mb + r)*HP + col] = pad ? 0.f : av;
        if (D) D[(mb + r)*HP + col] = pad ? 0.f : ((h > 0.f) ? 1.f : SLOPE);
      }
    }
  }
}

// ---- layer: [16 x 100] @ [100 x <=16]^T, single N tile, raw accumulator ----
__device__ __forceinline__ v8f out_layer(const float* Ain, const float* Wp,
                                         int l16, int kh)
{
  v8f acc = v8zero();
  const v2f* Ar = (const v2f*)(Ain + l16*HP + kh);
  const v2f* Br = (const v2f*)(Wp  + l16*HP + kh);
  #pragma unroll 1
  for (int k0 = 0; k0 < HP; k0 += 8) {
    const int ki = k0 >> 1;
    acc = wmma4(Ar[ki],     Br[ki],     acc);
    acc = wmma4(Ar[ki + 2], Br[ki + 2], acc);
  }
  return acc;
}

// ---------------- y0 = taskmap(0) — batch independent ----------------
__global__ void ngd_y0(const float* __restrict__ b1, const float* __restrict__ W2,
                       const float* __restrict__ b2, const float* __restrict__ W3,
                       const float* __restrict__ b3, float* __restrict__ y0)
{
  __shared__ float a1[HID], a2[HID];
  int t = threadIdx.x;
  if (t < HID) { float h = b1[t]; a1[t] = (h > 0.f) ? h : SLOPE * h; }
  __syncthreads();
  if (t < HID) {
    float s = b2[t];
    for (int k = 0; k < HID; ++k) s += W2[t*HID + k] * a1[k];
    a2[t] = (s > 0.f) ? s : SLOPE * s;
  }
  __syncthreads();
  if (t < NDIM) {
    float s = b3[t];
    for (int k = 0; k < HID; ++k) s += W3[t*HID + k] * a2[k];
    y0[t] = s;
  }
}

// ---------------- main fused kernel ----------------
__global__ void __launch_bounds__(256, 1)
ngd_main(const float* __restrict__ x,
         const float* __restrict__ W1, const float* __restrict__ b1,
         const float* __restrict__ W2, const float* __restrict__ b2,
         const float* __restrict__ W3, const float* __restrict__ b3,
         const float* __restrict__ V1, const float* __restrict__ c1,
         const float* __restrict__ V2, const float* __restrict__ c2,
         const float* __restrict__ V3, const float* __restrict__ c3,
         const float* __restrict__ y0g,
         float* __restrict__ out, int Bn)
{
  extern __shared__ float lds[];
  const int tid  = threadIdx.x;
  const int lane = tid & 31;
  const int wave = tid >> 5;
  const int l16  = lane & 15;
  const int kh   = (lane >> 4) << 1;   // K offset 0 / 2 (lane half)
  const int mb   = (lane >> 4) << 3;   // C/D row base 0 / 8 (lane half)

  // ---- stage zero-padded weights into LDS ----
  for (int i = tid; i < HR*NDIM; i += 256) {
    int r = i / NDIM, c = i - r*NDIM;
    lds[OFF_W1 + i] = (r < HID) ? W1[r*NDIM + c] : 0.f;
    lds[OFF_V1 + i] = (r < HID) ? V1[r*NDIM + c] : 0.f;
  }
  for (int i = tid; i < HR*HP; i += 256) {
    int r = i / HP, c = i - r*HP;
    bool in = (r < HID) && (c < HID);
    lds[OFF_W2 + i] = in ? W2[r*HID + c] : 0.f;
    lds[OFF_V2 + i] = in ? V2[r*HID + c] : 0.f;
  }
  for (int i = tid; i < 16*HP; i += 256) {
    int r = i / HP, c = i - r*HP;
    lds[OFF_W3 + i] = (r < NDIM && c < HID) ? W3[r*HID + c] : 0.f;
    lds[OFF_V3 + i] = (r == 0  && c < HID) ? V3[c]          : 0.f;
  }
  for (int i = tid; i < HR; i += 256) {
    lds[OFF_B1 + i] = (i < HID) ? b1[i] : 0.f;
    lds[OFF_B2 + i] = (i < HID) ? b2[i] : 0.f;
    lds[OFF_C1 + i] = (i < HID) ? c1[i] : 0.f;
    lds[OFF_C2 + i] = (i < HID) ? c2[i] : 0.f;
  }
  if (tid < NDIM) lds[OFF_B3 + tid] = b3[tid];
  if (tid == 0)   lds[OFF_C3] = c3[0];

  float* wv  = lds + OFF_WV + wave*WS;
  float* Act = wv + W_ACT;           // activations; later re-used as T' [16 x 104]
  float* D1  = wv + W_D1;
  float* D2  = wv + W_D2;
  float* Jb  = wv + W_J;
  float* Xb  = wv + W_X;
  float* Yd  = wv + W_YD;
  float* S0  = wv + W_S0;

  const long base = ((long)blockIdx.x * WAVES + wave) * ELEMS;
  for (int i = lane; i < ELEMS*NDIM; i += 32) {
    long g = base*NDIM + i;
    Xb[i] = (g < (long)Bn*NDIM) ? x[g] : 0.f;
  }
  __syncthreads();

  const float* sW1 = lds + OFF_W1; const float* sW2 = lds + OFF_W2;
  const float* sW3 = lds + OFF_W3; const float* sV1 = lds + OFF_V1;
  const float* sV2 = lds + OFF_V2; const float* sV3 = lds + OFF_V3;

  // ---- taskmap forward: x -> a1 (+d1) -> a2 (+d2) -> y ----
  layer_in (Xb,  sW1, lds + OFF_B1, Act, D1, l16, kh, mb);
  layer_hid(Act, sW2, lds + OFF_B2, Act, D2, l16, kh, mb);
  {
    v8f acc = out_layer(Act, sW3, l16, kh);   // W3 padded rows 8..15 = 0
    if (l16 < NDIM) {
      float bv  = lds[OFF_B3 + l16];
      float y0v = y0g[l16];
      #pragma unroll
      for (int r = 0; r < 8; ++r) {
        int m = mb + r;
        float yv = acc[r] + bv + Xb[m*NDIM + l16];   // residual +x
        Yd[m*NDIM + l16] = y0v - yv;                 // yd_hat = -(y - y0)
      }
    }
  }

  // ---- vel net: x -> g1 -> g2 -> s0 (Act buffer reused) ----
  layer_in (Xb,  sV1, lds + OFF_C1, Act, nullptr, l16, kh, mb);
  layer_hid(Act, sV2, lds + OFF_C2, Act, nullptr, l16, kh, mb);
  {
    v8f acc = out_layer(Act, sV3, l16, kh);   // only column 0 is live
    if (l16 == 0) {
      float c3v = lds[OFF_C3];
      #pragma unroll
      for (int r = 0; r < 8; ++r) S0[mb + r] = acc[r] + c3v;
    }
  }

  // ---- Jacobian: per pair of elements,
  //      T' = d2 .* (W2 @ (d1 .* W1)) stored transposed [n][h] (aliases Act),
  //      J  = W3 @ T' + I
  float* T = Act;                 // g2 is dead; reuse the activation buffer
  const int ecol = (l16 >> 3);    // which element of the pair this column serves
  const int nn   = l16 & 7;       // n index within element
  #pragma unroll 1
  for (int p = 0; p < ELEMS/2; ++p) {
    const int    e   = 2*p + ecol;
    const float* d1e = D1 + e*HP;
    const v2f*   d1r = (const v2f*)(d1e + kh);
    v8f acc[NT];
    #pragma unroll
    for (int t = 0; t < NT; ++t) acc[t] = v8zero();
    #pragma unroll 1
    for (int k0 = 0; k0 < HP; k0 += 8) {
      const int ki = k0 >> 1;
      const int k  = k0 + kh;
      v2f dA = d1r[ki];
      v2f dB = d1r[ki + 2];
      v2f b0, b1;                 // B[k][(e,n)] = d1_e[k] * W1[k][n]
      b0.x = dA.x * sW1[(k    )*NDIM + nn];
      b0.y = dA.y * sW1[(k + 1)*NDIM + nn];
      b1.x = dB.x * sW1[(k + 4)*NDIM + nn];
      b1.y = dB.y * sW1[(k + 5)*NDIM + nn];
      #pragma unroll
      for (int t = 0; t < NT; ++t) {
        const v2f* Ar = (const v2f*)(sW2 + (t*16 + l16)*HP + kh);
        acc[t] = wmma4(Ar[ki],     b0, acc[t]);
        acc[t] = wmma4(Ar[ki + 2], b1, acc[t]);
      }
    }
    // scale rows by d2_e and store transposed: T[l16][h]
    {
      const float* d2e  = D2 + e*HP;        // zero-padded rows >= 100
      float*       Trow = T + l16*HP;
      #pragma unroll
      for (int t = 0; t < NT; ++t) {
        #pragma unroll
        for (int r = 0; r < 8; ++r) {
          int h = t*16 + mb + r;
          if (h < HP) Trow[h] = acc[t][r] * d2e[h];
        }
      }
    }
    // J tile: A = W3p [16x104], B = T' (contiguous per lane); per-wave LDS in-order
    v8f j = v8zero();
    {
      const v2f* Ar = (const v2f*)(sW3 + l16*HP + kh);
      const v2f* Br = (const v2f*)(T   + l16*HP + kh);
      #pragma unroll 1
      for (int k0 = 0; k0 < HP; k0 += 8) {
        const int ki = k0 >> 1;
        j = wmma4(Ar[ki],     Br[ki],     j);
        j = wmma4(Ar[ki + 2], Br[ki + 2], j);
      }
    }
    if (mb == 0) {                           // rows o = 0..7 live in lanes 0..15
      #pragma unroll
      for (int r = 0; r < 8; ++r)
        Jb[e*64 + r*8 + nn] = j[r] + ((r == nn) ? 1.f : 0.f);
    }
  }

  // ---- per-element 8x8 solve (J z = yd), vel scaling, store ----
  if (lane < ELEMS) {
    const int e = lane;
    float M[8][8], rhs[8];
    #pragma unroll
    for (int i = 0; i < 8; ++i) {
      rhs[i] = Yd[e*NDIM + i];
      #pragma unroll
      for (int jx = 0; jx < 8; ++jx) M[i][jx] = Jb[e*64 + i*8 + jx];
    }
    #pragma unroll
    for (int k = 0; k < 8; ++k) {
      float inv = 1.0f / M[k][k];
      #pragma unroll
      for (int i = k + 1; i < 8; ++i) {
        float f = M[i][k] * inv;
        #pragma unroll
        for (int jx = k + 1; jx < 8; ++jx) M[i][jx] -= f * M[k][jx];
        rhs[i] -= f * rhs[k];
      }
    }
    float z[8];
    #pragma unroll
    for (int i = 7; i >= 0; --i) {
      float s = rhs[i];
      #pragma unroll
      for (int jx = i + 1; jx < 8; ++jx) s -= M[i][jx] * z[jx];
      z[i] = s / M[i][i];
    }
    if (base + e < Bn) {
      float s0 = S0[e];
      long  g  = (base + e) * NDIM;
      #pragma unroll
      for (int n = 0; n < 8; ++n) {
        float vel = __expf(s0 + Xb[e*NDIM + n]) + EPSV;
        out[g + n] = vel * z[n];
      }
    }
  }
}

extern "C" void kernel_launch(void* const* d_in, const int* in_sizes, int n_in,
                              void* d_out, int out_size, void* d_ws, size_t ws_size,
                              hipStream_t stream)
{
  (void)n_in; (void)out_size; (void)ws_size;
  const float* x  = (const float*)d_in[0];
  const float* W1 = (const float*)d_in[1];
  const float* b1 = (const float*)d_in[2];
  const float* W2 = (const float*)d_in[3];
  const float* b2 = (const float*)d_in[4];
  const float* W3 = (const float*)d_in[5];
  const float* b3 = (const float*)d_in[6];
  const float* V1 = (const float*)d_in[7];
  const float* c1 = (const float*)d_in[8];
  const float* V2 = (const float*)d_in[9];
  const float* c2 = (const float*)d_in[10];
  const float* V3 = (const float*)d_in[11];
  const float* c3 = (const float*)d_in[12];
  float* out = (float*)d_out;
  float* y0  = (float*)d_ws;

  const int Bn = in_sizes[0] / NDIM;
  ngd_y0<<<1, 128, 0, stream>>>(b1, W2, b2, W3, b3, y0);

  const int nwg = (Bn + WAVES*ELEMS - 1) / (WAVES*ELEMS);
  const size_t shmem = (size_t)LDS_FLOATS * sizeof(float);
  ngd_main<<<nwg, 256, shmem, stream>>>(x, W1, b1, W2, b2, W3, b3,
                                        V1, c1, V2, c2, V3, c3, y0, out, Bn);
}